// Sage_81389630259219
// MI455X (gfx1250) — compile-verified
//
#include <hip/hip_runtime.h>
#include <hip/hip_bf16.h>

// ---------------------------------------------------------------------------
// 2-layer GraphSAGE (mean aggregation), fp32, on MI455X (gfx1250).
//   layer: out = mean_agg(x_src) @ Wl + b + x_dst @ Wr    (+ReLU on layer 1)
// Dense parts use V_WMMA_F32_16X16X4_F32 (fp32 matrix cores, wave32).
// ---------------------------------------------------------------------------

typedef __attribute__((ext_vector_type(2))) float v2f;
typedef __attribute__((ext_vector_type(8))) float v8f;

#define N1C   81920
#define N2C   8192
#define DINC  128
#define DHC   256
#define DOUTC 128

// ---------------------------------------------------------------------------
// Edge scatter-add: one wave per edge, lanes stride the D-length feature row.
// agg[dst] += x[src]; cnt[dst] += 1
// ---------------------------------------------------------------------------
__global__ __launch_bounds__(256) void scatter_add_kernel(
    const float* __restrict__ xsrc, const int* __restrict__ src,
    const int* __restrict__ dst, float* __restrict__ agg,
    float* __restrict__ cnt, int E, int D)
{
    const int wavesPerBlock = blockDim.x >> 5;
    const int edge = blockIdx.x * wavesPerBlock + (threadIdx.x >> 5);
    const int lane = threadIdx.x & 31;
    if (edge >= E) return;

    const int s = src[edge];
    const int d = dst[edge];
    const float* __restrict__ xr = xsrc + (size_t)s * D;
    float* __restrict__ ar = agg + (size_t)d * D;

    #pragma unroll 4
    for (int c = lane; c < D; c += 32) {
        atomicAdd(ar + c, xr[c]);          // -> global_atomic_add_f32
    }
    if (lane == 0) atomicAdd(cnt + d, 1.0f);
}

// ---------------------------------------------------------------------------
// Fused SAGE dense stage:
//   out[m, :] = (agg[m,:]/max(cnt[m],1)) @ Wl + bias + xdst[m,:] @ Wr
// K = inner dim (128 or 256), N = output dim (256 or 128), fp32 WMMA.
// Block = 256 threads = 8 waves; one 16-row tile per block; each wave owns
// (N/16)/8 column tiles of 16. A operands staged in LDS, shared by all waves.
//
// V_WMMA_F32_16X16X4_F32 operand layouts (wave32):
//   A (16x4): lanes 0-15 -> {K=k, K=k+1}, lanes 16-31 -> {K=k+2, K=k+3}, M=lane&15
//   B (4x16): VGPR0: K = k + 2*(lane>>4); VGPR1: K+1;  N = lane&15
//   C (16x16): VGPR i: lanes 0-15 -> M=i, lanes 16-31 -> M=i+8; N = lane&15
// ---------------------------------------------------------------------------
template<int K, int N, bool RELU>
__global__ __launch_bounds__(256) void sage_gemm_kernel(
    const float* __restrict__ agg,  const float* __restrict__ cnt,
    const float* __restrict__ xdst,                       // [nrows, K]
    const float* __restrict__ wl,   const float* __restrict__ wr,  // [K, N]
    const float* __restrict__ bias,                       // [N]
    float* __restrict__ out, int nrows)                   // [nrows, N]
{
    __shared__ float Am[16][K + 4];   // mean operand
    __shared__ float Ad[16][K + 4];   // x_dst operand
    __shared__ float inv[16];

    const int tid  = threadIdx.x;
    const int row0 = blockIdx.x * 16;
    if (row0 >= nrows) return;        // block-uniform: EXEC stays all-1s

    // 1/max(cnt,1) per tile row
    if (tid < 16) {
        float cv = cnt[row0 + tid];
        inv[tid] = 1.0f / fmaxf(cv, 1.0f);
    }
    __syncthreads();

    // Cooperative LDS fill: 16*K elements each of mean and x_dst
    for (int idx = tid; idx < 16 * K; idx += 256) {
        const int r = idx / K;
        const int c = idx - r * K;
        Am[r][c] = agg [(size_t)(row0 + r) * K + c] * inv[r];
        Ad[r][c] = xdst[(size_t)(row0 + r) * K + c];
    }
    __syncthreads();

    const int wave = tid >> 5;
    const int lane = tid & 31;
    const int half = lane >> 4;       // selects K-pair within the 4-wide step
    const int lr   = lane & 15;       // A row / B,C column within the tile

    constexpr int TPW = (N / 16) / 8; // column tiles per wave (1 or 2)

    #pragma unroll
    for (int tt = 0; tt < TPW; ++tt) {
        const int n0 = (wave * TPW + tt) * 16;
        v8f c = {};

        #pragma unroll 4
        for (int k = 0; k < K; k += 4) {
            const int kk = k + half * 2;
            v2f a1, a2, bl, br;
            a1.x = Am[lr][kk];  a1.y = Am[lr][kk + 1];
            a2.x = Ad[lr][kk];  a2.y = Ad[lr][kk + 1];
            bl.x = wl[(size_t)kk * N + n0 + lr];
            bl.y = wl[(size_t)(kk + 1) * N + n0 + lr];
            br.x = wr[(size_t)kk * N + n0 + lr];
            br.y = wr[(size_t)(kk + 1) * N + n0 + lr];
            // D = A*B + C  (fp32 matrix core), both GEMM terms share one accum
            c = __builtin_amdgcn_wmma_f32_16x16x4_f32(
                    false, a1, false, bl, (short)0, c, false, false);
            c = __builtin_amdgcn_wmma_f32_16x16x4_f32(
                    false, a2, false, br, (short)0, c, false, false);
        }

        const int   col = n0 + lr;
        const float bv  = bias[col];
        #pragma unroll
        for (int i = 0; i < 8; ++i) {
            const int m = half * 8 + i;
            float v = c[i] + bv;
            if (RELU) v = fmaxf(v, 0.0f);
            out[(size_t)(row0 + m) * N + col] = v;
        }
    }
}

// ---------------------------------------------------------------------------
// Host-side orchestration (graph-capture safe: only async ops on `stream`).
// Input order: x, src1, dst1, src2, dst2, n1, n2, w1l, b1, w1r, w2l, b2, w2r
// ---------------------------------------------------------------------------
extern "C" void kernel_launch(void* const* d_in, const int* in_sizes, int n_in,
                              void* d_out, int out_size, void* d_ws, size_t ws_size,
                              hipStream_t stream) {
    const float* x    = (const float*)d_in[0];
    const int*   src1 = (const int*)  d_in[1];
    const int*   dst1 = (const int*)  d_in[2];
    const int*   src2 = (const int*)  d_in[3];
    const int*   dst2 = (const int*)  d_in[4];
    const float* w1l  = (const float*)d_in[7];
    const float* b1   = (const float*)d_in[8];
    const float* w1r  = (const float*)d_in[9];
    const float* w2l  = (const float*)d_in[10];
    const float* b2   = (const float*)d_in[11];
    const float* w2r  = (const float*)d_in[12];
    float* out = (float*)d_out;

    const int E1 = in_sizes[1];
    const int E2 = in_sizes[3];

    // Workspace carve-up (256B aligned)
    char* ws = (char*)d_ws;
    size_t off = 0;
    auto carve = [&](size_t bytes) {
        char* p = ws + off;
        off = (off + bytes + 255) & ~(size_t)255;
        return p;
    };
    float* agg1 = (float*)carve((size_t)N1C * DINC * sizeof(float)); // 42 MB
    float* cnt1 = (float*)carve((size_t)N1C * sizeof(float));
    float* h    = (float*)carve((size_t)N1C * DHC  * sizeof(float)); // 84 MB
    float* agg2 = (float*)carve((size_t)N2C * DHC  * sizeof(float)); //  8 MB
    float* cnt2 = (float*)carve((size_t)N2C * sizeof(float));
    (void)ws_size;

    // ---- Layer 1 ----
    hipMemsetAsync(agg1, 0, (size_t)N1C * DINC * sizeof(float), stream);
    hipMemsetAsync(cnt1, 0, (size_t)N1C * sizeof(float), stream);
    {
        const int wavesPerBlock = 8; // 256 threads
        const int blocks = (E1 + wavesPerBlock - 1) / wavesPerBlock;
        scatter_add_kernel<<<blocks, 256, 0, stream>>>(
            x, src1, dst1, agg1, cnt1, E1, DINC);
    }
    sage_gemm_kernel<DINC, DHC, true><<<N1C / 16, 256, 0, stream>>>(
        agg1, cnt1, /*xdst=*/x, w1l, w1r, b1, h, N1C);

    // ---- Layer 2 ----
    hipMemsetAsync(agg2, 0, (size_t)N2C * DHC * sizeof(float), stream);
    hipMemsetAsync(cnt2, 0, (size_t)N2C * sizeof(float), stream);
    {
        const int wavesPerBlock = 8;
        const int blocks = (E2 + wavesPerBlock - 1) / wavesPerBlock;
        scatter_add_kernel<<<blocks, 256, 0, stream>>>(
            h, src2, dst2, agg2, cnt2, E2, DHC);
    }
    sage_gemm_kernel<DHC, DOUTC, false><<<N2C / 16, 256, 0, stream>>>(
        agg2, cnt2, /*xdst=*/h, w2l, w2r, b2, out, N2C);

    (void)n_in; (void)out_size;
}